// TaxonomicLoss_62414464745492
// MI455X (gfx1250) — compile-verified
//
#include <hip/hip_runtime.h>
#include <hip/hip_bf16.h>
#include <math.h>

// Hierarchy constants from the reference (fixed).
#define N_NODES 20000
#define BATCH   4096
// LAYER_STARTS = 0, 100, 1100, 6100 ; sizes 100, 1000, 5000, 13900
#define PAR_END 6100   // nodes 0..6099 are potential parents (layers 0..2)

typedef __attribute__((ext_vector_type(2))) float v2f;
typedef __attribute__((ext_vector_type(8))) float v8f;

// Online softmax-stat merge: (m,s) <- combine((m,s),(m2,s2)), s = sum exp(x - m).
__device__ __forceinline__ void ms_combine(float& m, float& s, float m2, float s2) {
  float M = fmaxf(m, m2);
  if (M == -INFINITY) { m = M; s = 0.0f; return; }   // both empty
  s = s * __expf(m - M) + s2 * __expf(m2 - M);
  m = M;
}

__device__ __forceinline__ void ms_update(float& m, float& s, float x) {
  if (x > m) { s = s * __expf(m - x) + 1.0f; m = x; }
  else       { s += __expf(x - m); }
}

__device__ __forceinline__ void wave_ms(float& m, float& s) {
  #pragma unroll
  for (int off = 16; off; off >>= 1) {
    float m2 = __shfl_xor(m, off, 32);
    float s2 = __shfl_xor(s, off, 32);
    ms_combine(m, s, m2, s2);
  }
}

__device__ __forceinline__ float wave_sum32(float v) {
  #pragma unroll
  for (int off = 16; off; off >>= 1) v += __shfl_xor(v, off, 32);
  return v;
}

// One workgroup (1024 threads = 32 waves) per batch row. The entire 80 KB row
// is DMA'd into LDS with CDNA5 async-to-LDS (B128 bursts), then both the
// softmax statistics and the child pass (including the random parent gather)
// run out of LDS, so y_pred is read from HBM exactly once.
__global__ __launch_bounds__(1024)
void taxo_row_kernel(const float* __restrict__ y_pred,
                     const float* __restrict__ y_true,
                     const int*   __restrict__ parents,
                     const float* __restrict__ alpha,
                     float* __restrict__ loss_part)
{
  __shared__ __align__(16) float row_s[N_NODES];   // 80000 B: whole y_pred row
  __shared__ float red_s[32 * 12];                 // per-wave partials (reused twice)
  __shared__ float stat_s[8];                      // m[4], 1/S[4]

  const int tid  = threadIdx.x;
  const int lane = tid & 31;
  const int wv   = tid >> 5;
  const int row  = blockIdx.x;
  const float* rp = y_pred + (size_t)row * N_NODES;
  const float* ry = y_true + (size_t)row * N_NODES;

  // ---- Phase 1: async DMA the row into LDS, 16 B per lane per op
  //      (GVS mode: SGPR base + per-lane VGPR byte offset). 20000 f32 = 5000 xfers.
  for (int j4 = tid; j4 < (N_NODES / 4); j4 += 1024) {
    unsigned lds_off = (unsigned)(unsigned long long)(&row_s[j4 * 4]);
    unsigned goff    = (unsigned)(j4 * 16);
    asm volatile("global_load_async_to_lds_b128 %0, %1, %2"
                 :: "v"(lds_off), "v"(goff), "s"(rp) : "memory");
  }
  asm volatile("s_wait_asynccnt 0" ::: "memory");
  __syncthreads();

  // ---- Phase 2: per-segment (max, sum-exp) via online merge + wave32 shuffles
  float m0 = -INFINITY, s0 = 0.f, m1 = -INFINITY, s1 = 0.f;
  float m2 = -INFINITY, s2 = 0.f, m3 = -INFINITY, s3 = 0.f;
  for (int c = tid; c < N_NODES; c += 1024) {
    float x = row_s[c];
    if (c < 100)          ms_update(m0, s0, x);
    else if (c < 1100)    ms_update(m1, s1, x);
    else if (c < PAR_END) ms_update(m2, s2, x);
    else                  ms_update(m3, s3, x);
  }
  wave_ms(m0, s0); wave_ms(m1, s1); wave_ms(m2, s2); wave_ms(m3, s3);
  if (lane == 0) {
    red_s[wv*8+0] = m0; red_s[wv*8+1] = s0;
    red_s[wv*8+2] = m1; red_s[wv*8+3] = s1;
    red_s[wv*8+4] = m2; red_s[wv*8+5] = s2;
    red_s[wv*8+6] = m3; red_s[wv*8+7] = s3;
  }
  __syncthreads();
  if (tid == 0) {
    #pragma unroll
    for (int seg = 0; seg < 4; ++seg) {
      float M = -INFINITY, S = 0.f;
      for (int w = 0; w < 32; ++w) ms_combine(M, S, red_s[w*8 + seg*2], red_s[w*8 + seg*2 + 1]);
      stat_s[seg]     = M;
      stat_s[4 + seg] = 1.0f / S;
    }
  }
  __syncthreads();
  const float M0 = stat_s[0], M1 = stat_s[1], M2 = stat_s[2], M3 = stat_s[3];
  const float I0 = stat_s[4], I1 = stat_s[5], I2 = stat_s[6], I3 = stat_s[7];

  // ---- Phase 3: child pass. p from LDS row, pp gathered from LDS row,
  //      y_true streamed once from HBM.
  float z1=0,dp1=0,sy1=0,d1=0, z2=0,dp2=0,sy2=0,d2=0, z3=0,dp3=0,sy3=0,d3=0;
  for (int c = 100 + tid; c < N_NODES; c += 1024) {
    __builtin_prefetch(&ry[c + 1024], 0, 0);      // global_prefetch_b8
    float x  = row_s[c];
    float yt = ry[c];
    int   pi = parents[c];                        // global parent node index (< 6100)
    float mp, ip;
    if (pi < 100)       { mp = M0; ip = I0; }
    else if (pi < 1100) { mp = M1; ip = I1; }
    else                { mp = M2; ip = I2; }
    float pp = __expf(row_s[pi] - mp) * ip;       // parent softmax prob
    float mc, ic; int li;
    if (c < 1100)        { mc = M1; ic = I1; li = 0; }
    else if (c < PAR_END){ mc = M2; ic = I2; li = 1; }
    else                 { mc = M3; ic = I3; li = 2; }
    float p  = __expf(x - mc) * ic;               // child softmax prob
    float ez = __expf(p);                         // for Z2 = sum exp(p)
    float rd = fmaxf(p - pp, 0.0f);
    float yp = yt * p;
    if (li == 0)      { z1 += ez; dp1 += yp; sy1 += yt; d1 += rd; }
    else if (li == 1) { z2 += ez; dp2 += yp; sy2 += yt; d2 += rd; }
    else              { z3 += ez; dp3 += yp; sy3 += yt; d3 += rd; }
  }

  z1 = wave_sum32(z1); dp1 = wave_sum32(dp1); sy1 = wave_sum32(sy1); d1 = wave_sum32(d1);
  z2 = wave_sum32(z2); dp2 = wave_sum32(dp2); sy2 = wave_sum32(sy2); d2 = wave_sum32(d2);
  z3 = wave_sum32(z3); dp3 = wave_sum32(dp3); sy3 = wave_sum32(sy3); d3 = wave_sum32(d3);
  if (lane == 0) {
    float* r = &red_s[wv * 12];
    r[0]=z1; r[1]=dp1; r[2]=sy1; r[3]=d1;
    r[4]=z2; r[5]=dp2; r[6]=sy2; r[7]=d2;
    r[8]=z3; r[9]=dp3; r[10]=sy3; r[11]=d3;
  }
  __syncthreads();
  if (tid == 0) {
    float acc[12];
    #pragma unroll
    for (int k = 0; k < 12; ++k) {
      float t = 0.f;
      for (int w = 0; w < 32; ++w) t += red_s[w*12 + k];
      acc[k] = t;
    }
    float a1 = alpha[1], a2 = alpha[2], a3 = alpha[3];
    // cce = log(Z2)*sum(y) - dot(y,p) ; d = D / n_layer
    float loss =
        a1 * (logf(acc[0]) * acc[2]  - acc[1]  + acc[3]  * (1.0f / 1000.0f)) +
        a2 * (logf(acc[4]) * acc[6]  - acc[5]  + acc[7]  * (1.0f / 5000.0f)) +
        a3 * (logf(acc[8]) * acc[10] - acc[9]  + acc[11] * (1.0f / 13900.0f));
    loss_part[row] = loss;
  }
}

// Final 4096 -> 1 reduction on the matrix pipe: C += ones(16x4) x B(4x16).
// With A all ones, D[m][n] = sum_k B[k][n] (independent of m), so summing
// VGPR0 of C across the wave counts every input exactly twice. Element
// placement inside B is irrelevant for the total, so each lane loads one
// contiguous v2f pair (global_load_b64).
__global__ __launch_bounds__(32)
void taxo_final_reduce(const float* __restrict__ part, float* __restrict__ out)
{
  const int lane = threadIdx.x;
  const v2f* __restrict__ part2 = (const v2f*)part;
  v2f a; a[0] = 1.0f; a[1] = 1.0f;
  v8f c = {};
  for (int iter = 0; iter < (BATCH / 64); ++iter) {
    v2f b = part2[iter * 32 + lane];   // 64 contiguous floats per iteration
    c = __builtin_amdgcn_wmma_f32_16x16x4_f32(
        /*neg_a=*/false, a, /*neg_b=*/false, b,
        /*c_mod=*/(short)0, c, /*reuse_a=*/false, /*reuse_b=*/false);
  }
  float v = wave_sum32(c[0]);          // = 2 * total
  if (lane == 0) out[0] = v * 0.5f * (1.0f / (float)BATCH);
}

extern "C" void kernel_launch(void* const* d_in, const int* in_sizes, int n_in,
                              void* d_out, int out_size, void* d_ws, size_t ws_size,
                              hipStream_t stream) {
  const float* y_pred  = (const float*)d_in[0];
  const float* y_true  = (const float*)d_in[1];
  const int*   parents = (const int*)d_in[2];
  const float* alpha   = (const float*)d_in[3];
  float* loss_part = (float*)d_ws;   // BATCH floats of scratch

  taxo_row_kernel<<<BATCH, 1024, 0, stream>>>(y_pred, y_true, parents, alpha, loss_part);
  taxo_final_reduce<<<1, 32, 0, stream>>>(loss_part, (float*)d_out);
}